// Edge_CNN_50036368998511
// MI455X (gfx1250) — compile-verified
//
#include <hip/hip_runtime.h>
#include <math.h>

typedef float v2f __attribute__((ext_vector_type(2)));
typedef float v8f __attribute__((ext_vector_type(8)));
typedef unsigned int v4u __attribute__((ext_vector_type(4)));
typedef int v4i __attribute__((ext_vector_type(4)));
typedef int v8i __attribute__((ext_vector_type(8)));

#define BB 32
#define NN 1024
#define KNBR 10

__device__ __forceinline__ v8f wmma_f32_16x16x4(v2f a, v2f b, v8f c) {
  // 8 args: (neg_a, A, neg_b, B, c_mod, C, reuse_a, reuse_b)
  return __builtin_amdgcn_wmma_f32_16x16x4_f32(false, a, false, b, (short)0, c,
                                               false, false);
}

// ---------------------------------------------------------------------------
// sq[b,n] = sum_d x[b,n,d]^2
// ---------------------------------------------------------------------------
template <int D>
__global__ void sq_kernel(const float* __restrict__ X, float* __restrict__ sq) {
  int i = blockIdx.x * blockDim.x + threadIdx.x;
  if (i >= BB * NN) return;
  const float* x = X + (size_t)i * D;
  float s = 0.f;
#pragma unroll
  for (int d = 0; d < D; ++d) s += x[d] * x[d];
  sq[i] = s;
}

// ---------------------------------------------------------------------------
// kNN: one wave per (batch, 16-query tile).
// dist GEMM via V_WMMA_F32_16X16X4_F32, 16x1024 distance stripe in LDS,
// then 16 lanes each insertion-sort a top-10 list over their row.
// LDS layout sdist[col*16 + row] (64 KB exactly) -> conflict-free row scan.
// ---------------------------------------------------------------------------
template <int D>
__global__ __launch_bounds__(32) void knn_kernel(const float* __restrict__ X,
                                                 const float* __restrict__ sq,
                                                 int* __restrict__ knn_out) {
  __shared__ float sdist[16 * NN];

  const int b = blockIdx.x >> 6;          // NN/16 = 64 tiles per batch
  const int n0 = (blockIdx.x & 63) << 4;
  const int lane = threadIdx.x;
  const int r = lane & 15;                // A-row / B-col / C-col index
  const int hi = lane >> 4;               // K half select

  const float* Xb = X + (size_t)b * NN * D;
  constexpr int CH = (D + 3) / 4;

  // A-tile (queries n0..n0+15) resident in VGPRs: lane r holds row r,
  // vgpr pair c holds K = 4c+2*hi, 4c+2*hi+1 (zero-padded when D%4 != 0).
  v2f a[CH];
#pragma unroll
  for (int c = 0; c < CH; ++c) {
    int k = 4 * c + 2 * hi;
    a[c].x = (k < D) ? Xb[(size_t)(n0 + r) * D + k] : 0.f;
    a[c].y = (k + 1 < D) ? Xb[(size_t)(n0 + r) * D + k + 1] : 0.f;
  }
  float srow[8];
#pragma unroll
  for (int v = 0; v < 8; ++v) srow[v] = sq[b * NN + n0 + v + 8 * hi];

  for (int m0 = 0; m0 < NN; m0 += 16) {
    v8f acc = {};
#pragma unroll
    for (int c = 0; c < CH; ++c) {
      int k = 4 * c + 2 * hi;
      v2f bm;
      bm.x = (k < D) ? Xb[(size_t)(m0 + r) * D + k] : 0.f;
      bm.y = (k + 1 < D) ? Xb[(size_t)(m0 + r) * D + k + 1] : 0.f;
      acc = wmma_f32_16x16x4(a[c], bm, acc);  // acc += A * B  (x . x^T)
    }
    float sqm = sq[b * NN + m0 + r];  // column node |x_m|^2
#pragma unroll
    for (int v = 0; v < 8; ++v) {
      // C layout: vgpr v, lanes 0-15 -> row v, lanes 16-31 -> row v+8; col = r
      sdist[(m0 + r) * 16 + (v + 8 * hi)] = srow[v] + sqm - 2.0f * acc[v];
    }
  }
  __syncthreads();

  if (lane < 16) {
    float bd[KNBR];
    int bi[KNBR];
#pragma unroll
    for (int t = 0; t < KNBR; ++t) { bd[t] = 3.402823e38f; bi[t] = 0; }
    for (int j = 0; j < NN; ++j) {
      float v = sdist[j * 16 + lane];
      if (v < bd[KNBR - 1]) {
        int p = KNBR - 1;
        while (p > 0 && v < bd[p - 1]) {
          bd[p] = bd[p - 1];
          bi[p] = bi[p - 1];
          --p;
        }
        bd[p] = v;
        bi[p] = j;
      }
    }
    int* o = knn_out + (size_t)(b * NN + n0 + lane) * KNBR;
#pragma unroll
    for (int t = 0; t < KNBR; ++t) o[t] = bi[t];
  }
}

// ---------------------------------------------------------------------------
// Weight re-pack:  edge@W = [xi, xj-xi]@[Wt;Wb] = xj@Wb + xi@(Wt-Wb)
// so with A'' = [xj | xi] rows we need W'' = [Wb ; Wt - Wb].
// ---------------------------------------------------------------------------
template <int DIN, int DOUT>
__global__ void wprep_kernel(const float* __restrict__ W,
                             float* __restrict__ Wp) {
  int i = blockIdx.x * blockDim.x + threadIdx.x;
  if (i >= 2 * DIN * DOUT) return;
  int k = i / DOUT, o = i % DOUT;
  float v;
  if (k < DIN)
    v = W[(size_t)(DIN + k) * DOUT + o];                            // Wb
  else
    v = W[(size_t)(k - DIN) * DOUT + o] - W[(size_t)k * DOUT + o];  // Wt-Wb
  Wp[i] = v;
}

// ---------------------------------------------------------------------------
// EdgeConv, one wave per node. The 10-neighbor row gather is done by the
// Tensor Data Mover in GATHER mode: 16 x DIN rows (16-bit row indices from
// the kNN list) DMA'd into LDS, with TDM pad_interval/pad_amount inserting a
// DIN-dword hole after every DIN-dword row -> row stride 2*DIN. The wave
// fills the holes with x_i (disjoint bytes) while the DMA runs, then a
// single s_wait_tensorcnt orders the DMA before the WMMA loop.
// ---------------------------------------------------------------------------
template <int DIN, int DOUT>
__global__ __launch_bounds__(32) void edgeconv_tdm_kernel(
    const float* __restrict__ X, const int* __restrict__ knn_in,
    const float* __restrict__ Wp, const float* __restrict__ bias,
    float* __restrict__ Y) {
  constexpr int E = 2 * DIN;
  __shared__ __align__(64) float eM[16][E];  // [row][0..DIN)=x_j, [DIN..E)=x_i

  const int node = blockIdx.x;  // 0 .. B*N-1
  const int b = node >> 10;
  const int lane = threadIdx.x;
  const int r = lane & 15;
  const int hi = lane >> 4;

  const float* Xb = X + (size_t)b * NN * DIN;
  const float* xn = X + (size_t)node * DIN;
  const int* kn = knn_in + (size_t)node * KNBR;

  // 10 neighbor ids (uniform across the wave) + 6 repeats for rows 10..15
  int id[16];
#pragma unroll
  for (int t = 0; t < 16; ++t) id[t] = kn[t < KNBR ? t : KNBR - 1];

  // ---- Tensor DMA descriptor (D#), gather mode, 16-bit indices ----
  v4i g2, g3;
  g2.x = id[0] | (id[1] << 16);
  g2.y = id[2] | (id[3] << 16);
  g2.z = id[4] | (id[5] << 16);
  g2.w = id[6] | (id[7] << 16);
  g3.x = id[8] | (id[9] << 16);
  g3.y = id[10] | (id[11] << 16);
  g3.z = id[12] | (id[13] << 16);
  g3.w = id[14] | (id[15] << 16);

  unsigned long long ga = (unsigned long long)(uintptr_t)Xb;
  unsigned int ldsa = (unsigned int)(uintptr_t)&eM[0][0];  // LDS byte offset

  v4u g0;
  g0.x = 1u | (1u << 31);  // count=1 | gather_index_size=16b | gather_mode=1
  g0.y = ldsa;                                  // lds_addr
  g0.z = (unsigned int)ga;                      // global_addr[31:0]
  g0.w = (unsigned int)((ga >> 32) & 0x01FFFFFFu) | (2u << 30);  // [56:32]|type=2

  constexpr unsigned PADI = (DIN == 2) ? 0u : (DIN == 64) ? 5u : 6u;  // 2^(c+1) dw
  constexpr unsigned PADA = (unsigned)(DIN - 1);                      // c+1 dw
  v8i g1;
  g1[0] = (int)((2u << 16)            // data_size = 4 bytes
                | (1u << 20)          // pad_enable
                | (PADI << 22)        // pad_interval: DIN dwords of data
                | (PADA << 25));      // pad_amount:   DIN dwords of hole
  g1[1] = (int)(((unsigned)DIN) << 16);              // tensor_dim0 (row width)
  g1[2] = (int)(((unsigned)NN) << 16);               // tensor_dim1 = N rows
  g1[3] = (int)(((unsigned)DIN) << 16);              // tile_dim0 = DIN
  g1[4] = 16;                                        // tile_dim1 = 16 indices
  g1[5] = DIN;                                       // tensor_dim0_stride lo32
  g1[6] = 0;
  g1[7] = 0;

  v8i g4 = {};  // unused trailing SGPR group (6-arg toolchain form): zeros
  __builtin_amdgcn_tensor_load_to_lds(g0, g1, g2, g3, g4, 0);  // TENSORcnt++

  // Fill x_i into columns [DIN, 2*DIN) of all 16 rows while the DMA streams
  // the x_j halves (disjoint LDS bytes -> no race with the TDM writes).
  for (int c = lane; c < DIN; c += 32) {
    float v = xn[c];
#pragma unroll
    for (int row = 0; row < 16; ++row) eM[row][DIN + c] = v;
  }
  __builtin_amdgcn_s_wait_tensorcnt(0);
  __syncthreads();

#pragma unroll 1
  for (int t = 0; t < DOUT; t += 16) {
    v8f acc = {};
#pragma unroll
    for (int k0 = 0; k0 < E; k0 += 4) {
      int k = k0 + 2 * hi;
      v2f av, wv;
      av.x = eM[r][k];
      av.y = eM[r][k + 1];
      wv.x = Wp[(size_t)k * DOUT + t + r];
      wv.y = Wp[(size_t)(k + 1) * DOUT + t + r];
      acc = wmma_f32_16x16x4(av, wv, acc);
    }
    // max over neighbor rows 0..9 of this 16x16 msg tile (column = r)
    float m;
    if (hi == 0) {
      m = acc[0];
#pragma unroll
      for (int v = 1; v < 8; ++v) m = fmaxf(m, acc[v]);  // rows 0..7
    } else {
      m = fmaxf(acc[0], acc[1]);  // rows 8,9 (rows 10..15 are padding)
    }
    m = fmaxf(m, __shfl_xor(m, 16, 32));
    if (hi == 0) {
      float val = m + bias[t + r];
      Y[(size_t)node * DOUT + t + r] = fmaxf(val, 0.f);  // ReLU after max
    }
  }
}

// ---------------------------------------------------------------------------
// PairNorm (deterministic two-stage reduction, no float atomics):
// msn = sum(y^2)/(B*N) - sum_c mean_c^2 ;  out = (y - mean_c) * rsqrt(eps+msn)
// ---------------------------------------------------------------------------
template <int C>
__global__ void pn_partial(const float* __restrict__ Y, float* __restrict__ psum,
                           float* __restrict__ psq) {
  __shared__ float red[256];
  const int tid = threadIdx.x;  // blockDim.x == C
  const int rows = (BB * NN) / gridDim.x;
  const int r0 = blockIdx.x * rows;
  float s = 0.f, q = 0.f;
  for (int rr = 0; rr < rows; ++rr) {
    float v = Y[(size_t)(r0 + rr) * C + tid];
    s += v;
    q += v * v;
  }
  psum[blockIdx.x * C + tid] = s;
  red[tid] = q;
  __syncthreads();
  for (int st = blockDim.x / 2; st > 0; st >>= 1) {
    if (tid < st) red[tid] += red[tid + st];
    __syncthreads();
  }
  if (tid == 0) psq[blockIdx.x] = red[0];
}

template <int C>
__global__ void pn_finalize(const float* __restrict__ psum,
                            const float* __restrict__ psq,
                            float* __restrict__ stats, int nblk) {
  __shared__ float red[256];
  const int tid = threadIdx.x;  // blockDim.x == C
  float s = 0.f;
  for (int i = 0; i < nblk; ++i) s += psum[i * C + tid];
  float mean = s * (1.0f / (float)(BB * NN));
  stats[tid] = mean;
  red[tid] = mean * mean;
  __syncthreads();
  for (int st = blockDim.x / 2; st > 0; st >>= 1) {
    if (tid < st) red[tid] += red[tid + st];
    __syncthreads();
  }
  if (tid == 0) {
    float q = 0.f;
    for (int i = 0; i < nblk; ++i) q += psq[i];
    float msn = q * (1.0f / (float)(BB * NN)) - red[0];
    stats[C] = 1.0f / sqrtf(1e-5f + msn);
  }
}

template <int C>
__global__ void pn_apply(float* __restrict__ Y, const float* __restrict__ stats) {
  int i = blockIdx.x * blockDim.x + threadIdx.x;
  if (i >= BB * NN * C) return;
  int c = i & (C - 1);
  Y[i] = (Y[i] - stats[c]) * stats[C];
}

// ---------------------------------------------------------------------------
// g[b,c] = max_n h3[b,n,c]
// ---------------------------------------------------------------------------
__global__ void gmax_kernel(const float* __restrict__ H, float* __restrict__ G) {
  int i = blockIdx.x * blockDim.x + threadIdx.x;
  if (i >= BB * 256) return;
  int b = i >> 8, c = i & 255;
  const float* p = H + (size_t)b * NN * 256 + c;
  float m = -3.402823e38f;
  for (int n = 0; n < NN; ++n) m = fmaxf(m, p[(size_t)n * 256]);
  G[i] = m;
}

// ---------------------------------------------------------------------------
// out = relu(g @ Wl1 + bl1) @ Wl2 + bl2   (tiny: one block per batch row)
// ---------------------------------------------------------------------------
__global__ __launch_bounds__(64) void mlp_kernel(
    const float* __restrict__ G, const float* __restrict__ Wl1,
    const float* __restrict__ bl1, const float* __restrict__ Wl2,
    const float* __restrict__ bl2, float* __restrict__ out) {
  __shared__ float gb[256];
  __shared__ float hb[64];
  const int b = blockIdx.x, t = threadIdx.x;
  for (int c = t; c < 256; c += 64) gb[c] = G[b * 256 + c];
  __syncthreads();
  float s = bl1[t];
  for (int k = 0; k < 256; ++k) s += gb[k] * Wl1[k * 64 + t];
  hb[t] = fmaxf(s, 0.f);
  __syncthreads();
  if (t < 2) {
    float o = bl2[t];
    for (int k = 0; k < 64; ++k) o += hb[k] * Wl2[k * 2 + t];
    out[b * 2 + t] = o;
  }
}

// ---------------------------------------------------------------------------
extern "C" void kernel_launch(void* const* d_in, const int* in_sizes, int n_in,
                              void* d_out, int out_size, void* d_ws,
                              size_t ws_size, hipStream_t stream) {
  const float* pos = (const float*)d_in[0];
  const float* W1 = (const float*)d_in[1];
  const float* b1 = (const float*)d_in[2];
  const float* W2 = (const float*)d_in[3];
  const float* b2 = (const float*)d_in[4];
  const float* W3 = (const float*)d_in[5];
  const float* b3 = (const float*)d_in[6];
  const float* Wl1 = (const float*)d_in[7];
  const float* bl1 = (const float*)d_in[8];
  const float* Wl2 = (const float*)d_in[9];
  const float* bl2 = (const float*)d_in[10];
  float* out = (float*)d_out;

  char* base = (char*)d_ws;
  size_t off = 0;
  auto carve = [&](size_t bytes) -> void* {
    void* p = base + off;
    off = (off + bytes + 255) & ~(size_t)255;
    return p;
  };
  const size_t SZFEAT = (size_t)BB * NN * 256 * sizeof(float);  // 32 MB
  const int RB = 128;

  float* bufA = (float*)carve(SZFEAT);
  float* bufB = (float*)carve(SZFEAT);
  float* sqb = (float*)carve((size_t)BB * NN * sizeof(float));
  int* knnb = (int*)carve((size_t)BB * NN * KNBR * sizeof(int));
  float* psum = (float*)carve((size_t)RB * 256 * sizeof(float));
  float* psq = (float*)carve((size_t)RB * sizeof(float));
  float* stats = (float*)carve(257 * sizeof(float));
  float* g = (float*)carve((size_t)BB * 256 * sizeof(float));
  float* wp = (float*)carve((size_t)256 * 256 * sizeof(float));  // W'' scratch

  const int nodes = BB * NN;

  // ---- Layer 1: pos (d=2) -> bufA (64 ch) ----
  sq_kernel<2><<<(nodes + 255) / 256, 256, 0, stream>>>(pos, sqb);
  knn_kernel<2><<<BB * (NN / 16), 32, 0, stream>>>(pos, sqb, knnb);
  wprep_kernel<2, 64><<<(4 * 64 + 255) / 256, 256, 0, stream>>>(W1, wp);
  edgeconv_tdm_kernel<2, 64><<<nodes, 32, 0, stream>>>(pos, knnb, wp, b1, bufA);
  pn_partial<64><<<RB, 64, 0, stream>>>(bufA, psum, psq);
  pn_finalize<64><<<1, 64, 0, stream>>>(psum, psq, stats, RB);
  pn_apply<64><<<(nodes * 64 + 255) / 256, 256, 0, stream>>>(bufA, stats);

  // ---- Layer 2: bufA (64) -> bufB (128 ch) ----
  sq_kernel<64><<<(nodes + 255) / 256, 256, 0, stream>>>(bufA, sqb);
  knn_kernel<64><<<BB * (NN / 16), 32, 0, stream>>>(bufA, sqb, knnb);
  wprep_kernel<64, 128><<<(128 * 128 + 255) / 256, 256, 0, stream>>>(W2, wp);
  edgeconv_tdm_kernel<64, 128><<<nodes, 32, 0, stream>>>(bufA, knnb, wp, b2,
                                                         bufB);
  pn_partial<128><<<RB, 128, 0, stream>>>(bufB, psum, psq);
  pn_finalize<128><<<1, 128, 0, stream>>>(psum, psq, stats, RB);
  pn_apply<128><<<(nodes * 128 + 255) / 256, 256, 0, stream>>>(bufB, stats);

  // ---- Layer 3: bufB (128) -> bufA (256 ch) ----
  sq_kernel<128><<<(nodes + 255) / 256, 256, 0, stream>>>(bufB, sqb);
  knn_kernel<128><<<BB * (NN / 16), 32, 0, stream>>>(bufB, sqb, knnb);
  wprep_kernel<128, 256><<<(256 * 256 + 255) / 256, 256, 0, stream>>>(W3, wp);
  edgeconv_tdm_kernel<128, 256><<<nodes, 32, 0, stream>>>(bufB, knnb, wp, b3,
                                                          bufA);
  pn_partial<256><<<RB, 256, 0, stream>>>(bufA, psum, psq);
  pn_finalize<256><<<1, 256, 0, stream>>>(psum, psq, stats, RB);
  pn_apply<256><<<(nodes * 256 + 255) / 256, 256, 0, stream>>>(bufA, stats);

  // ---- Global max pool + MLP head ----
  gmax_kernel<<<(BB * 256 + 255) / 256, 256, 0, stream>>>(bufA, g);
  mlp_kernel<<<BB, 64, 0, stream>>>(g, Wl1, bl1, Wl2, bl2, out);
}